// QuadraticsSpline_25580825215457
// MI455X (gfx1250) — compile-verified
//
#include <hip/hip_runtime.h>
#include <math.h>

#define M_ROWS   65536
#define S_IN     48
#define S_OUT    1056
#define N_SPL    32
#define K_BINS   16
#define BLK_ROWS 32
#define AS_STRIDE 49
#define Y_STRIDE  1060   /* == 4 mod 8 -> D-fragment half-writes hit disjoint banks */

typedef __attribute__((ext_vector_type(2))) float v2f;
typedef __attribute__((ext_vector_type(8))) float v8f;

__global__ __launch_bounds__(256)
void qspline_fused_kernel(const float* __restrict__ c,
                          const float* __restrict__ z,
                          const float* __restrict__ W,
                          const float* __restrict__ bias,
                          float* __restrict__ out_x,
                          float* __restrict__ out_ld)
{
    extern __shared__ float smem[];
    float* As = smem;                              // BLK_ROWS x AS_STRIDE
    float* Ys = smem + BLK_ROWS * AS_STRIDE;       // BLK_ROWS x Y_STRIDE

    const int tid  = threadIdx.x;
    const int lane = tid & 31;
    const int wave = tid >> 5;
    const int row0 = blockIdx.x * BLK_ROWS;

    // ---- Stage A = [z2 | c] (32 x 48) into LDS ----
    for (int i = tid; i < BLK_ROWS * S_IN; i += 256) {
        int r = i / S_IN;
        int k = i - r * S_IN;
        float v = (k < 32) ? z[(size_t)(row0 + r) * 64 + 32 + k]
                           : c[(size_t)(row0 + r) * 16 + (k - 32)];
        As[r * AS_STRIDE + k] = v;
    }
    __syncthreads();

    // ---- GEMM: Y = A @ W + b via V_WMMA_F32_16X16X4_F32 ----
    const int rt      = wave >> 2;       // 0..1 : row tile (16 rows)
    const int halfsel = lane >> 4;       // 0 or 1
    const int l15     = lane & 15;
    const int arow    = rt * 16 + l15;

    // A fragments: VGPR0 = K(0|2), VGPR1 = K(1|3); lanes16-31 take K+2
    v2f afrag[12];
#pragma unroll
    for (int kk = 0; kk < 12; ++kk) {
        int k = kk * 4 + halfsel * 2;
        afrag[kk].x = As[arow * AS_STRIDE + k];
        afrag[kk].y = As[arow * AS_STRIDE + k + 1];
    }

    for (int ct = (wave & 3); ct < 66; ct += 4) {
        int col = ct * 16 + l15;
        v8f acc = {0.f, 0.f, 0.f, 0.f, 0.f, 0.f, 0.f, 0.f};
#pragma unroll
        for (int kk = 0; kk < 12; ++kk) {
            int kb = kk * 4 + halfsel * 2;   // B mirrors A: lanes16-31 hold K+2
            v2f bfrag;
            bfrag.x = W[(size_t)kb * S_OUT + col];
            bfrag.y = W[(size_t)(kb + 1) * S_OUT + col];
            acc = __builtin_amdgcn_wmma_f32_16x16x4_f32(
                false, afrag[kk], false, bfrag, (short)0, acc, false, false);
        }
        float bv = bias[col];
#pragma unroll
        for (int v = 0; v < 8; ++v) {
            int r = rt * 16 + v + halfsel * 8;   // D: VGPR v -> row v (+8 for hi lanes)
            Ys[r * Y_STRIDE + col] = acc[v] + bv;
        }
    }
    __syncthreads();

    // ---- x[:, :32] = z2 (from staged A) ----
    for (int i = tid; i < BLK_ROWS * 32; i += 256) {
        int r = i >> 5, j = i & 31;
        out_x[(size_t)(row0 + r) * 64 + j] = As[r * AS_STRIDE + j];
    }

    // ---- Quadratic spline: one (row) per wave-iteration, lane = n ----
    for (int it = 0; it < 4; ++it) {
        const int r = it * 8 + wave;              // 0..31
        const size_t grow = (size_t)row0 + r;
        const int n = lane;
        const float* sv = &Ys[r * Y_STRIDE + n * (2 * K_BINS + 1)];

        // raw params: heights sv[0..16], widths sv[17..32]
        float uw[K_BINS];
        float h[K_BINS + 1];
#pragma unroll
        for (int i = 0; i < K_BINS; ++i) uw[i] = sv[K_BINS + 1 + i];
#pragma unroll
        for (int j = 0; j < K_BINS + 1; ++j) {
            float x = sv[j];                       // softplus + 0.001
            h[j] = fmaxf(x, 0.f) + log1pf(expf(-fabsf(x))) + 0.001f;
        }

        // widths = 0.001 + (1 - 0.001*16) * softmax(uw)
        float mx = uw[0];
#pragma unroll
        for (int i = 1; i < K_BINS; ++i) mx = fmaxf(mx, uw[i]);
        float se = 0.f;
        float width[K_BINS];
#pragma unroll
        for (int i = 0; i < K_BINS; ++i) { width[i] = expf(uw[i] - mx); se += width[i]; }
        float inv_se = 1.f / se;
#pragma unroll
        for (int i = 0; i < K_BINS; ++i)
            width[i] = 0.001f + (1.f - 0.001f * K_BINS) * width[i] * inv_se;

        // area and normalized heights  H_i = 0.001 + 0.999 * h_i / area
        float area = 0.f;
#pragma unroll
        for (int i = 0; i < K_BINS; ++i)
            area += (h[i] + h[i + 1]) * 0.5f * width[i];
        float inv_area = 1.f / area;

        const float input = z[grow * 64 + n];      // z1

        float loc_left = 0.f, cdf_left = 0.f;
        float g_loc = 0.f, g_wid = width[0], g_cdf = 0.f;
        float g_hl = 0.001f + 0.999f * h[0] * inv_area;
        float g_hr = 0.001f + 0.999f * h[1] * inv_area;
        bool done = false;
#pragma unroll
        for (int i = 0; i < K_BINS; ++i) {
            float wid = width[i];
            float Hl = 0.001f + 0.999f * h[i] * inv_area;
            float Hr = 0.001f + 0.999f * h[i + 1] * inv_area;
            float loc_right = (i == K_BINS - 1) ? 1.0f : (loc_left + wid);
            float thresh    = (i == K_BINS - 1) ? (1.0f + 1e-6f) : loc_right;
            bool sel = (!done) && ((input < thresh) || (i == K_BINS - 1));
            g_loc = sel ? loc_left : g_loc;
            g_wid = sel ? wid      : g_wid;
            g_cdf = sel ? cdf_left : g_cdf;
            g_hl  = sel ? Hl       : g_hl;
            g_hr  = sel ? Hr       : g_hr;
            done = done || sel;
            cdf_left += (Hl + Hr) * 0.5f * wid;
            loc_left = loc_right;
        }

        float alpha = (input - g_loc) / g_wid;
        float aq = 0.5f * (g_hr - g_hl) * g_wid;
        float bq = g_hl * g_wid;
        float outv = aq * alpha * alpha + bq * alpha + g_cdf;
        outv = fminf(fmaxf(outv, 0.f), 1.f);
        float ld = logf(alpha * (g_hr - g_hl) + g_hl);

        out_x[grow * 64 + 32 + n] = outv;

        // wave reduction of log_det over n
#pragma unroll
        for (int off = 16; off >= 1; off >>= 1)
            ld += __shfl_xor(ld, off, 32);
        if (lane == 0) out_ld[grow] = ld;
    }
}

extern "C" void kernel_launch(void* const* d_in, const int* in_sizes, int n_in,
                              void* d_out, int out_size, void* d_ws, size_t ws_size,
                              hipStream_t stream) {
    (void)in_sizes; (void)n_in; (void)out_size; (void)d_ws; (void)ws_size;
    const float* c    = (const float*)d_in[0];
    const float* z    = (const float*)d_in[1];
    const float* W    = (const float*)d_in[2];
    const float* bias = (const float*)d_in[3];
    float* out_x  = (float*)d_out;
    float* out_ld = out_x + (size_t)M_ROWS * 64;

    const size_t lds_bytes = (size_t)(BLK_ROWS * AS_STRIDE + BLK_ROWS * Y_STRIDE) * sizeof(float);
    dim3 grid(M_ROWS / BLK_ROWS);
    dim3 block(256);
    qspline_fused_kernel<<<grid, block, lds_bytes, stream>>>(c, z, W, bias, out_x, out_ld);
}